// SimpleHOGModule_40020505264237
// MI455X (gfx1250) — compile-verified
//
#include <hip/hip_runtime.h>
#include <hip/hip_bf16.h>
#include <math.h>

// ---------------------------------------------------------------------------
// HOG-style Sobel-binning operator for MI455X (gfx1250).
//   input : x (64^3 f32)            [d_in[0]]   (weight d_in[1] is a fixed
//                                                central-difference kernel,
//                                                hardcoded here)
//   output: (1, 8, 8, 78, 78, 78) f32 = 30,371,328 elements (121.5 MB)
//
// Strategy (store-bandwidth bound, ~5.2us floor @ 23.3 TB/s):
//   1) per thread: 4 consecutive flat voxels; compute gradients, mag/theta/phi,
//      bin indices + weights, merge duplicate bins in-register.
//   2) stream zeros to all 64 bin planes with CDNA5 async b128 stores
//      (LDS -> global, ASYNCcnt), s_wait_asynccnt 0.
//   3) overwrite the <=3 nonzero (bin,voxel) cells with plain b32 stores
//      (same-thread same-address stores are ordered; cells are uniquely owned).
// ---------------------------------------------------------------------------

#define DVOL 64
#define OD   78
#define NVOX (OD * OD * OD)       // 474552 (divisible by 4)
#define VPT  4                    // voxels per thread
#define BLOCK 256

#if defined(__HIP_DEVICE_COMPILE__) && defined(__gfx1250__) && \
    __has_builtin(__builtin_amdgcn_global_store_async_from_lds_b128)
#define USE_ASYNC_ZSTORE 1
#else
#define USE_ASYNC_ZSTORE 0
#endif

// native 4 x i32 vector (what the async builtins traffic in)
typedef int v4i __attribute__((ext_vector_type(4)));
typedef __attribute__((address_space(1))) v4i gbl_v4i_t;
typedef __attribute__((address_space(3))) v4i lds_v4i_t;

__device__ __forceinline__ float ldx(const float* __restrict__ X, int z, int y, int x) {
  if ((unsigned)z < (unsigned)DVOL && (unsigned)y < (unsigned)DVOL &&
      (unsigned)x < (unsigned)DVOL)
    return X[(((z << 6) + y) << 6) + x];
  return 0.0f;
}

__global__ __launch_bounds__(BLOCK) void hog_sobel_bins_kernel(
    const float* __restrict__ X, float* __restrict__ out) {
  __shared__ float4 zslab[BLOCK];
  const int tid = threadIdx.x;
  zslab[tid] = make_float4(0.f, 0.f, 0.f, 0.f);
  __syncthreads();  // LDS zeros visible before async engine reads them

  const int v0 = (blockIdx.x * BLOCK + tid) * VPT;
  if (v0 >= NVOX) return;

  int   b0v[VPT], b1v[VPT], b2v[VPT];
  float w0v[VPT], w1v[VPT], w2v[VPT];
  bool  s1v[VPT], s2v[VPT];

  const float PI_F = 3.14159265358979323846f;          // fl(pi)
  const float EPSF = 2.220446049250313e-16f;           // sys.float_info.epsilon

#pragma unroll
  for (int j = 0; j < VPT; ++j) {
    const int v = v0 + j;
    const int z = v / (OD * OD);
    const int r = v - z * (OD * OD);
    const int y = r / OD;
    const int x = r - y * OD;

    // central differences (conv w/ pad 8, kernel 3 -> taps at -8..-6)
    const float sz = ldx(X, z - 6, y - 7, x - 7) - ldx(X, z - 8, y - 7, x - 7);
    const float sy = ldx(X, z - 7, y - 6, x - 7) - ldx(X, z - 7, y - 8, x - 7);
    const float sx = ldx(X, z - 7, y - 7, x - 6) - ldx(X, z - 7, y - 7, x - 8);

    const float mag   = sqrtf(sz * sz + sy * sy + sx * sx);
    const float theta = atan2f(sy, sx);
    const float phi   = acosf(sz / (mag + EPSF));

    // faithful to reference: (angle / pi) * 8  (x8 is exact)
    const float t_raw = theta / PI_F * 8.0f;
    const float p_raw = phi   / PI_F * 8.0f;

    const float tf = t_raw - truncf(t_raw);   // torch.frac semantics
    const float pf = p_raw - truncf(p_raw);

    const int i0 = ((int)floorf(t_raw)) & 7;  // == numpy %8 for [-8,8]
    const int i1 = ((int)ceilf (t_raw)) & 7;
    const int i2 = ((int)floorf(p_raw)) & 7;
    const int i3 = ((int)ceilf (p_raw)) & 7;

    const float f0 = fabsf(tf), f1 = fabsf(1.0f - tf);
    const float f2 = fabsf(pf), f3 = fabsf(1.0f - pf);
    float w0 = (f0 * f2) * mag;   // -> (i0, i2)
    float w1 = (f0 * f3) * mag;   // -> (i0, i3)
    float w2 = (f1 * f2) * mag;   // -> (i1, i2)

    const int b0 = i0 * 8 + i2, b1 = i0 * 8 + i3, b2 = i1 * 8 + i2;
    bool s1 = (b1 != b0);
    if (!s1) w0 += w1;            // merge duplicate deposits (reference adds)
    bool s2 = true;
    if (b2 == b0)            { w0 += w2; s2 = false; }
    else if (s1 && b2 == b1) { w1 += w2; s2 = false; }

    b0v[j] = b0; b1v[j] = b1; b2v[j] = b2;
    w0v[j] = w0; w1v[j] = w1; w2v[j] = w2;
    s1v[j] = s1; s2v[j] = s2;
  }

  // ---- phase 1: zero-fill all 64 bin planes for these 4 voxels (b128) ----
#if USE_ASYNC_ZSTORE
  // LDS source: this thread's 16B zero slot. Generic->AS3 via low-32 offset
  // (flat-aperture rule: LDS generic addr[31:0] == LDS offset).
  lds_v4i_t* lsrc =
      (lds_v4i_t*)(unsigned int)(unsigned long long)(void*)&zslab[tid];
  float* gp = out + v0;
  for (int bin = 0; bin < 64; ++bin) {
    __builtin_amdgcn_global_store_async_from_lds_b128(
        (gbl_v4i_t*)(unsigned long long)(void*)gp, lsrc, 0, 0);
    gp += NVOX;
  }
  asm volatile("" ::: "memory");
#if __has_builtin(__builtin_amdgcn_s_wait_asynccnt)
  __builtin_amdgcn_s_wait_asynccnt(0);
#else
  asm volatile("s_wait_asynccnt 0" ::: "memory");
#endif
  asm volatile("" ::: "memory");
#else
  {
    const float4 z4 = make_float4(0.f, 0.f, 0.f, 0.f);
    float* gp = out + v0;
    for (int bin = 0; bin < 64; ++bin) {
      *(float4*)gp = z4;
      gp += NVOX;
    }
  }
#endif

  // ---- phase 2: overwrite the <=3 nonzero cells per voxel ----
#pragma unroll
  for (int j = 0; j < VPT; ++j) {
    const int v = v0 + j;
    out[(size_t)b0v[j] * NVOX + v] = w0v[j];
    if (s1v[j]) out[(size_t)b1v[j] * NVOX + v] = w1v[j];
    if (s2v[j]) out[(size_t)b2v[j] * NVOX + v] = w2v[j];
  }
}

extern "C" void kernel_launch(void* const* d_in, const int* in_sizes, int n_in,
                              void* d_out, int out_size, void* d_ws, size_t ws_size,
                              hipStream_t stream) {
  (void)in_sizes; (void)n_in; (void)out_size; (void)d_ws; (void)ws_size;
  const float* x = (const float*)d_in[0];  // d_in[1] (weight) is a fixed
                                           // central-difference kernel: hardcoded
  float* out = (float*)d_out;
  const int nthreads = NVOX / VPT;                       // 118638
  const int nblocks  = (nthreads + BLOCK - 1) / BLOCK;   // 464
  hog_sobel_bins_kernel<<<nblocks, BLOCK, 0, stream>>>(x, out);
}